// NBVLoss_88450556494508
// MI455X (gfx1250) — compile-verified
//
#include <hip/hip_runtime.h>

// BCE full-sum reduction for B=32768, V=1024 f32 inputs (out = scalar f32).
// Memory-bound: 256 MiB streamed @ 23.3 TB/s => ~11.5us floor.
//   - 128-bit non-temporal loads, 2x unrolled grid-stride (4 loads in flight)
//   - t in {0,1}: select x = t ? p : 1-p, ONE native log2 per element,
//     clamp in log2 domain (-100/ln2), scale by -ln2 once at the end.
//   - wave32 reduction via V_WMMA_F32_16X16X4_F32 (ones-vector trick)
//   - LDS block combine, deterministic 2-kernel grid reduction (no atomics)

typedef float v2f __attribute__((ext_vector_type(2)));
typedef float v4f __attribute__((ext_vector_type(4)));
typedef float v8f __attribute__((ext_vector_type(8)));

#define LOG2_CLAMP  (-144.26950408889634f)   // -100 / ln(2)
#define NEG_LN2     (-0.6931471805599453f)

// Per-element contribution in log2 domain:
//   bce = -max(log(x), -100)  where x = t ? p : (1-p)
//       = -ln2 * max(log2(x), -100/ln2)
__device__ __forceinline__ float bce_log2(float p, float t) {
    float x = (t != 0.0f) ? p : (1.0f - p);
    return fmaxf(__log2f(x), LOG2_CLAMP);
}

// Wave32 sum via WMMA f32 16x16x4.
// A (16x4): VGPR0 = acc, VGPR1 = 0 -> A[m][0]=acc(lane m), A[m][2]=acc(lane m+16)
// B (4x16) = ones => D[m][n] = acc(m) + acc(m+16) for every n.
// Per-lane sum of the 8 D VGPRs = rows 0..7 (lanes 0-15) / rows 8..15 (16-31);
// lane0 + lane16 = full wave sum.  REQUIRES full EXEC.
__device__ __forceinline__ float wave_reduce_wmma(float acc) {
    v2f a; a[0] = acc;  a[1] = 0.0f;
    v2f b; b[0] = 1.0f; b[1] = 1.0f;
    v8f c = {};
    v8f d = __builtin_amdgcn_wmma_f32_16x16x4_f32(
        false, a, false, b, (short)0, c, false, false);
    float s = ((d[0] + d[1]) + (d[2] + d[3])) + ((d[4] + d[5]) + (d[6] + d[7]));
    return __shfl(s, 0, 32) + __shfl(s, 16, 32);
}

__device__ __forceinline__ float block_reduce(float wsum, float* smem) {
    const int lane = threadIdx.x & 31;
    const int wave = threadIdx.x >> 5;
    if (lane == 0) smem[wave] = wsum;
    __syncthreads();
    float v = 0.0f;
    if (wave == 0) {
        v = (lane < (int)(blockDim.x >> 5)) ? smem[lane] : 0.0f;
        v += __shfl_down(v, 16, 32);
        v += __shfl_down(v,  8, 32);
        v += __shfl_down(v,  4, 32);
        v += __shfl_down(v,  2, 32);
        v += __shfl_down(v,  1, 32);
    }
    return v;   // valid in wave 0 lane 0
}

__global__ void __launch_bounds__(256)
bce_partial_kernel(const float* __restrict__ pred,
                   const float* __restrict__ targ,
                   float* __restrict__ block_partials,
                   long long n4, long long n) {
    __shared__ float smem[8];
    const long long tid    = (long long)blockIdx.x * blockDim.x + threadIdx.x;
    const long long stride = (long long)gridDim.x * blockDim.x;
    const v4f* __restrict__ p4 = (const v4f*)pred;
    const v4f* __restrict__ t4 = (const v4f*)targ;

    float acc = 0.0f;
    long long i = tid;
    // 2x unrolled: 4 independent 128-bit NT loads in flight per iteration.
    for (; i + stride < n4; i += 2 * stride) {
        v4f p0 = __builtin_nontemporal_load(p4 + i);
        v4f t0 = __builtin_nontemporal_load(t4 + i);
        v4f p1 = __builtin_nontemporal_load(p4 + i + stride);
        v4f t1 = __builtin_nontemporal_load(t4 + i + stride);
        acc += bce_log2(p0[0], t0[0]);
        acc += bce_log2(p0[1], t0[1]);
        acc += bce_log2(p0[2], t0[2]);
        acc += bce_log2(p0[3], t0[3]);
        acc += bce_log2(p1[0], t1[0]);
        acc += bce_log2(p1[1], t1[1]);
        acc += bce_log2(p1[2], t1[2]);
        acc += bce_log2(p1[3], t1[3]);
    }
    if (i < n4) {
        v4f p0 = __builtin_nontemporal_load(p4 + i);
        v4f t0 = __builtin_nontemporal_load(t4 + i);
        acc += bce_log2(p0[0], t0[0]);
        acc += bce_log2(p0[1], t0[1]);
        acc += bce_log2(p0[2], t0[2]);
        acc += bce_log2(p0[3], t0[3]);
    }
    // Scalar tail (n not divisible by 4).
    for (long long j = 4 * n4 + tid; j < n; j += stride) {
        acc += bce_log2(pred[j], targ[j]);
    }

    float wsum = wave_reduce_wmma(acc);      // full EXEC: no divergence above
    float v = block_reduce(wsum, smem);
    if (threadIdx.x == 0) block_partials[blockIdx.x] = v;   // log2 domain
}

__global__ void __launch_bounds__(256)
bce_final_kernel(const float* __restrict__ partials,
                 float* __restrict__ out, int n) {
    __shared__ float smem[8];
    float acc = 0.0f;
    for (int i = threadIdx.x; i < n; i += (int)blockDim.x) acc += partials[i];

    float wsum = wave_reduce_wmma(acc);      // all 256 threads active
    float v = block_reduce(wsum, smem);
    if (threadIdx.x == 0) out[0] = NEG_LN2 * v;   // lambdas are both 1.0
}

extern "C" void kernel_launch(void* const* d_in, const int* in_sizes, int n_in,
                              void* d_out, int out_size, void* d_ws, size_t ws_size,
                              hipStream_t stream) {
    const float* pred = (const float*)d_in[0];
    const float* targ = (const float*)d_in[1];
    float* out      = (float*)d_out;
    float* partials = (float*)d_ws;

    const long long n  = (long long)in_sizes[0];   // 33,554,432
    const long long n4 = n >> 2;

    const int block = 256;   // 8 wave32s
    const int grid  = 2048;  // 524,288 threads -> 16 x 128-bit loads each

    bce_partial_kernel<<<grid, block, 0, stream>>>(pred, targ, partials, n4, n);
    bce_final_kernel<<<1, block, 0, stream>>>(partials, out, grid);
}